// GLCMFeatures_79242146612003
// MI455X (gfx1250) — compile-verified
//
#include <hip/hip_runtime.h>
#include <hip/hip_bf16.h>
#include <math.h>

// ---------------------------------------------------------------------------
// GLCM texture features for 32 x (256x256) images, 8 offsets, 256 levels.
//   out[b, f, h, w] = feats[b][f]  (broadcast), f in [0,64)
//
// MI455X notes:
//  * 320KB LDS/WGP -> full 256x256 u32 GLCM histogram (256KB) in LDS,
//    incremented with ds_add_u32 (no global atomics).
//  * Cross-wave reduction of 32x16 partial-sum matrix done with 8 chained
//    V_WMMA_F32_16X16X4_F32 (A = ones => exact fp32 column sums).
//  * 512MB broadcast output written with nontemporal b128 stores.
// ---------------------------------------------------------------------------

typedef __attribute__((ext_vector_type(2))) float v2f;
typedef __attribute__((ext_vector_type(4))) float v4f;
typedef __attribute__((ext_vector_type(8))) float v8f;

// OFFSETS from the reference (angles interpreted in *radians* by np.sin/cos):
//   a=0:    (0,1)  (0,2)
//   a=90:   (1,0)  (2,-1)
//   a=-45:  (-1,1) (-2,1)
//   a=-135: (0,-1) (0,-2)
__constant__ int OFFS[8][2] = {
    {0, 1}, {0, 2}, {1, 0}, {2, -1}, {-1, 1}, {-2, 1}, {0, -1}, {0, -2}};

#define H 256
#define W 256
#define NPIX (H * W)
#define NIMG 32
#define NOFF 8
#define LEVELS 256
#define HIST_BINS (LEVELS * LEVELS)

// ---------------------------------------------------------------- min / max
__global__ __launch_bounds__(256) void glcm_minmax_k(
    const float* __restrict__ x, float* __restrict__ mmx) {
  const int b = blockIdx.x;
  const float* p = x + (size_t)b * NPIX;
  float mn = 1e30f, mx = -1e30f;
  for (int i = threadIdx.x; i < NPIX; i += 256) {
    float v = p[i];
    mn = fminf(mn, v);
    mx = fmaxf(mx, v);
  }
  for (int o = 16; o > 0; o >>= 1) {
    mn = fminf(mn, __shfl_xor(mn, o, 32));
    mx = fmaxf(mx, __shfl_xor(mx, o, 32));
  }
  __shared__ float smn[8], smx[8];
  const int wid = threadIdx.x >> 5, lane = threadIdx.x & 31;
  if (lane == 0) { smn[wid] = mn; smx[wid] = mx; }
  __syncthreads();
  if (threadIdx.x == 0) {
    for (int w = 1; w < 8; ++w) { mn = fminf(mn, smn[w]); mx = fmaxf(mx, smx[w]); }
    mmx[2 * b + 0] = mn;
    mmx[2 * b + 1] = 255.0f / (mx - mn + 1e-8f);  // scale
  }
}

// ----------------------------------------------------------------- quantize
__global__ __launch_bounds__(256) void glcm_quant_k(
    const float* __restrict__ x, const float* __restrict__ mmx,
    unsigned char* __restrict__ q) {
  const size_t i = (size_t)blockIdx.x * 256 + threadIdx.x;  // < NIMG*NPIX
  const int b = (int)(i >> 16);
  const float mn = mmx[2 * b + 0];
  const float sc = mmx[2 * b + 1];
  float qa = floorf((x[i] - mn) * sc);
  qa = fminf(fmaxf(qa, 0.0f), 255.0f);
  q[i] = (unsigned char)qa;
}

// ------------------------------------------- per (image, offset) GLCM + feats
// blockIdx.x = img*8 + off ; 1024 threads = 32 waves
// dynamic LDS: [ u32 hist[65536] | float red[32][16] ]
__global__ __launch_bounds__(1024, 1) void glcm_hist_feat_k(
    const unsigned char* __restrict__ qall, float* __restrict__ feats) {
  extern __shared__ unsigned char smem[];
  unsigned int* lhist = (unsigned int*)smem;
  float(*red)[16] = (float(*)[16])(smem + HIST_BINS * 4);

  const int tid = threadIdx.x;
  const int img = blockIdx.x >> 3;
  const int off = blockIdx.x & 7;
  const int dr = OFFS[off][0], dc = OFFS[off][1];
  const unsigned char* q = qall + (size_t)img * NPIX;

  // -- zero histogram
  for (int i = tid; i < HIST_BINS; i += 1024) lhist[i] = 0u;
  __syncthreads();

  // -- histogram of co-occurring pairs (valid region), ds_add_u32 atomics
  const int r0 = max(0, -dr), r1 = H - max(0, dr);
  const int c0 = max(0, -dc), c1 = W - max(0, dc);
  const int rh = r1 - r0;
  for (int idx = tid; idx < rh * 256; idx += 1024) {
    const int rr = r0 + (idx >> 8);
    const int cc = c0 + (idx & 255);
    if (cc < c1) {
      const int va = q[rr * W + cc];
      const int vb = q[(rr + dr) * W + (cc + dc)];
      atomicAdd(&lhist[(va << 8) + vb], 1u);
    }
  }
  __syncthreads();

  // -- single-pass feature scan over symmetrized, normalized GLCM.
  //    S = 2 * #pairs is known analytically -> one pass suffices.
  const float invS = 1.0f / fmaxf((float)(2 * rh * (c1 - c0)), 1.0f);
  float a[10];
#pragma unroll
  for (int k = 0; k < 10; ++k) a[k] = 0.0f;
  // a: 0 contrast, 1 dissim, 2 homog, 3 asm, 4 entropy,
  //    5 mi, 6 mj, 7 mii, 8 mjj, 9 mij
  for (int e = tid; e < HIST_BINS; e += 1024) {
    const int i = e >> 8, j = e & 255;
    const float g = (float)(lhist[e] + lhist[(j << 8) + i]);
    const float p = g * invS;
    const float fi = (float)i, fj = (float)j;
    const float d = fi - fj;
    const float d2 = d * d;
    a[0] += p * d2;
    a[1] += p * fabsf(d);
    a[2] += p / (1.0f + d2);
    a[3] += p * p;
    const float pe = p + 1e-8f;
    a[4] -= pe * __log2f(pe);
    a[5] += fi * p;
    a[6] += fj * p;
    a[7] += fi * fi * p;
    a[8] += fj * fj * p;
    a[9] += fi * fj * p;
  }

  // -- intra-wave butterfly reduction (wave32)
#pragma unroll
  for (int k = 0; k < 10; ++k) {
#pragma unroll
    for (int o = 16; o > 0; o >>= 1) a[k] += __shfl_xor(a[k], o, 32);
  }
  const int wid = tid >> 5, lane = tid & 31;
  if (lane == 0) {
#pragma unroll
    for (int k = 0; k < 10; ++k) red[wid][k] = a[k];
#pragma unroll
    for (int k = 10; k < 16; ++k) red[wid][k] = 0.0f;
  }
  __syncthreads();

  // -- cross-wave reduction: 8 chained V_WMMA_F32_16X16X4_F32 with A = ones.
  //    D[m,n] += sum_k B[k,n]; with A all-ones the result is the exact fp32
  //    column sum of red[0..31][n], independent of the B K<->lane mapping as
  //    long as each (half-wave, vgpr) slot carries a distinct row — which the
  //    indexing below guarantees.
  if (tid < 32) {  // wave 0, EXEC all ones
    v8f acc = {};
    const int half = lane >> 4;   // 0 | 1
    const int n = lane & 15;      // column (feature) index
#pragma unroll
    for (int c = 0; c < 8; ++c) {
      v2f am; am.x = 1.0f; am.y = 1.0f;
      v2f bm;
      bm.x = red[c * 4 + half * 2 + 0][n];
      bm.y = red[c * 4 + half * 2 + 1][n];
      acc = __builtin_amdgcn_wmma_f32_16x16x4_f32(
          /*neg_a=*/false, am, /*neg_b=*/false, bm,
          /*c_mod=*/(short)0, acc, /*reuse_a=*/false, /*reuse_b=*/false);
    }
    // row M=0 of D lives in acc[0], lanes 0..15 hold N=lane
    if (lane < 16) red[0][lane] = acc[0];
  }
  __syncthreads();

  // -- derive the 8 reference features
  if (tid == 0) {
    const float contrast = red[0][0];
    const float dissim   = red[0][1];
    const float homog    = red[0][2];
    const float asum     = red[0][3];
    const float entropy  = red[0][4];
    const float mi  = red[0][5], mj  = red[0][6];
    const float mii = red[0][7], mjj = red[0][8], mij = red[0][9];

    const float var_i = mii - mi * mi;
    const float var_j = mjj - mj * mj;
    const float std_i = sqrtf(fmaxf(var_i, 0.0f));
    const float std_j = sqrtf(fmaxf(var_j, 0.0f));
    const float cov = mij - mi * mj;
    const float denom = std_i * std_j;
    const float corr = (denom < 1e-15f) ? 1.0f : cov / fmaxf(denom, 1e-15f);

    float* f = feats + (size_t)img * 64 + (size_t)off * 8;
    f[0] = contrast;
    f[1] = dissim;
    f[2] = homog;
    f[3] = sqrtf(asum);  // energy
    f[4] = corr;
    f[5] = asum;
    f[6] = entropy;
    f[7] = var_i;        // variance (mean over rows)
  }
}

// --------------------------------------------------------------- broadcast
// out[b, f, :, :] = feats[b*64+f]; 512MB stream -> nontemporal b128 stores
__global__ __launch_bounds__(256) void glcm_bcast_k(
    const float* __restrict__ feats, float* __restrict__ out) {
  const size_t i = (size_t)blockIdx.x * 256 + threadIdx.x;  // v4f index
  const size_t e = i << 2;                                  // element index
  const float v = feats[e >> 16];                           // (b*64+f)
  v4f w;
  w.x = v; w.y = v; w.z = v; w.w = v;
  __builtin_nontemporal_store(w, ((v4f*)out) + i);
}

// ---------------------------------------------------------------------------
extern "C" void kernel_launch(void* const* d_in, const int* in_sizes, int n_in,
                              void* d_out, int out_size, void* d_ws,
                              size_t ws_size, hipStream_t stream) {
  const float* x = (const float*)d_in[0];  // (32,1,256,256) fp32
  float* out = (float*)d_out;              // (32,64,256,256) fp32

  // workspace layout
  float* feats = (float*)d_ws;                                   // 32*64 f32
  float* mmx = (float*)((char*)d_ws + 8192);                     // 32*2  f32
  unsigned char* q = (unsigned char*)((char*)d_ws + 16384);      // 32*65536 u8

  glcm_minmax_k<<<NIMG, 256, 0, stream>>>(x, mmx);

  glcm_quant_k<<<(NIMG * NPIX) / 256, 256, 0, stream>>>(x, mmx, q);

  const size_t lds_bytes = (size_t)HIST_BINS * 4 + 32 * 16 * sizeof(float);
  glcm_hist_feat_k<<<NIMG * NOFF, 1024, lds_bytes, stream>>>(q, feats);

  const int nvec = out_size / 4;  // 128-bit stores
  glcm_bcast_k<<<(nvec + 255) / 256, 256, 0, stream>>>(feats, out);
}